// Mamba_Flow_align_11690900980491
// MI455X (gfx1250) — compile-verified
//
#include <hip/hip_runtime.h>
#include <math.h>

// ---------------------------------------------------------------------------
// MI455X (gfx1250) implementation.
// All dense matmuls (implicit-GEMM 3x3 convs, attention q/k/v/o projections,
// Mamba in/xproj/dt/out projections, 1x1 conv) run on V_WMMA_F32_16X16X4_F32.
// Each wave accumulates a 16-row x 128-col strip (8 accumulator tiles) so one
// activation fragment load feeds 8 WMMAs. Out-of-range N columns load clamped
// weight rows and skip the store (no EXEC manipulation in the inner loop).
// ---------------------------------------------------------------------------

typedef __attribute__((ext_vector_type(2))) float v2f;
typedef __attribute__((ext_vector_type(8))) float v8f;

#define DEVI __device__ __forceinline__

constexpr int kC  = 128;
constexpr int kN  = 2;
constexpr int kH  = 64;
constexpr int kW  = 64;
constexpr int kHW = kH * kW;          // 4096
constexpr int kT  = 7;
constexpr int kNH = 8;
constexpr int kDI = 128;
constexpr int kDS = 16;
constexpr int kL  = kT * kHW;         // 28672
constexpr size_t kNCHW = (size_t)kN * kC * kHW; // 1,048,576

#define CDIV(a, b) (((a) + (b) - 1) / (b))

DEVI v8f wmma4(v2f a, v2f b, v8f c) {
  // D = A(16x4) * B(4x16) + C, fp32
  return __builtin_amdgcn_wmma_f32_16x16x4_f32(false, a, false, b, (short)0, c,
                                               false, false);
}

DEVI float actf(float v, int act) {
  if (act == 1) return v >= 0.f ? v : 0.1f * v;        // lrelu(0.1)
  if (act == 2) return v > 20.f ? v : log1pf(expf(v)); // softplus
  return v;
}

DEVI int clampi(int v, int lo, int hi) { return v < lo ? lo : (v > hi ? hi : v); }

// ---------------------------------------------------------------------------
// WMMA GEMM:  Y[r, co] = act( sum_k X[r, k] * Wt[co, k] + bias[co] )
// One wave: 16 rows x (NT*16) columns, NT accumulators; the X fragment is
// loaded once per k-step and reused by NT WMMAs. R % 64 == 0.
// ---------------------------------------------------------------------------
template <int NT>
__global__ void gemm_xw(const float* __restrict__ X, int ldx, int K,
                        const float* __restrict__ Wt, int ldw,
                        const float* __restrict__ bias, float* __restrict__ Y,
                        int ldy, int R, int Co, int act) {
  int wave = threadIdx.x >> 5;
  int lane = threadIdx.x & 31;
  int half = lane >> 4;
  int i16  = lane & 15;
  int r0 = (blockIdx.x * 4 + wave) * 16;
  if (r0 >= R) return; // wave-uniform
  int cobase = blockIdx.y * (NT * 16);
  const float* xrow = X + (size_t)(r0 + i16) * ldx + half * 2;
  const float* wrow[NT];
  int coL[NT];
#pragma unroll
  for (int t = 0; t < NT; ++t) {
    int co = cobase + t * 16 + i16;
    coL[t] = co;
    int cs = co < Co ? co : (Co - 1); // clamped: garbage cols never stored
    wrow[t] = Wt + (size_t)cs * ldw + half * 2;
  }
  v8f zero = {0.f, 0.f, 0.f, 0.f, 0.f, 0.f, 0.f, 0.f};
  v8f acc[NT];
#pragma unroll
  for (int t = 0; t < NT; ++t) acc[t] = zero;
  for (int kb = 0; kb < K; kb += 4) {
    v2f a;
    a.x = xrow[kb];
    a.y = xrow[kb + 1];
#pragma unroll
    for (int t = 0; t < NT; ++t) {
      v2f b;
      b.x = wrow[t][kb];
      b.y = wrow[t][kb + 1];
      acc[t] = wmma4(a, b, acc[t]);
    }
  }
#pragma unroll
  for (int t = 0; t < NT; ++t) {
    int co = coL[t];
    if (co < Co) {
      float bv = bias ? bias[co] : 0.f;
#pragma unroll
      for (int r = 0; r < 8; ++r) {
        int row = r0 + r + half * 8;
        Y[(size_t)row * ldy + co] = actf(acc[t][r] + bv, act);
      }
    }
  }
}

// ---------------------------------------------------------------------------
// Implicit-GEMM 3x3 conv (pad=1), N=2, H=W=64, NCHW, Cout=128.
// Weights pre-packed as Wp[tap][co][ci] (contiguous ci -> b64 A loads).
// One wave: 16 pixels x 128 output channels (8 tiles); the im2col fragment is
// loaded once per k-step (clamped address * {0,1} pad mask, no EXEC games)
// and reused by 8 WMMAs. Input channels may come from two concatenated
// tensors (conv13).  out = resid + act(conv + bias)
// ---------------------------------------------------------------------------
__global__ void conv3x3_wmma_k(const float* __restrict__ in1, int C1,
                               const float* __restrict__ in2, int C2,
                               const float* __restrict__ Wp,
                               const float* __restrict__ bias,
                               const float* __restrict__ resid,
                               float* __restrict__ out, int act) {
  const int Cin = C1 + C2;
  int wave = threadIdx.x >> 5;
  int lane = threadIdx.x & 31;
  int half = lane >> 4;
  int i16  = lane & 15;
  int pt  = blockIdx.x * 4 + wave;  // 512 pixel tiles (b, y, x0/16)
  int b   = pt >> 8;
  int rem = pt & 255;
  int y   = rem >> 2;
  int x0  = (rem & 3) << 4;
  int px  = x0 + i16;

  __builtin_prefetch(in1 + ((size_t)b * C1) * kHW + y * 64 + x0, 0, 1);

  v8f zero = {0.f, 0.f, 0.f, 0.f, 0.f, 0.f, 0.f, 0.f};
  v8f acc[8];
#pragma unroll
  for (int t = 0; t < 8; ++t) acc[t] = zero;

  for (int tap = 0; tap < 9; ++tap) {
    int ky = tap / 3, kx = tap % 3;
    int yy = y + ky - 1;
    int xx = px + kx - 1;
    float msk = (yy >= 0 && yy < 64 && xx >= 0 && xx < 64) ? 1.f : 0.f;
    int sp = clampi(yy, 0, 63) * 64 + clampi(xx, 0, 63);
    const float* wtap = Wp + (size_t)tap * 128 * Cin + half * 2;
    for (int cb = 0; cb < Cin; cb += 4) {
      int ci = cb + half * 2;
      // fragment row pair ci, ci+1 (same tensor: C1 is even, ci is even)
      const float* src = (ci < C1)
                             ? (in1 + ((size_t)b * C1 + ci) * kHW)
                             : (in2 + ((size_t)b * C2 + (ci - C1)) * kHW);
      v2f bf;
      bf.x = src[sp] * msk;
      bf.y = src[kHW + sp] * msk;
#pragma unroll
      for (int t = 0; t < 8; ++t) {
        const float* wp = wtap + (size_t)(t * 16 + i16) * Cin + cb;
        v2f a;
        a.x = wp[0];
        a.y = wp[1];
        acc[t] = wmma4(a, bf, acc[t]);
      }
    }
  }
  int spo = y * 64 + x0 + i16;
#pragma unroll
  for (int t = 0; t < 8; ++t) {
#pragma unroll
    for (int r = 0; r < 8; ++r) {
      int co = t * 16 + r + half * 8;
      float v = actf(acc[t][r] + bias[co], act);
      if (resid) v += resid[((size_t)b * 128 + co) * kHW + spo];
      out[((size_t)b * 128 + co) * kHW + spo] = v;
    }
  }
}

// pack W[co][ci][3][3] -> Wp[tap][co][ci]
__global__ void wpack9(const float* __restrict__ w, float* __restrict__ wp,
                       int Cin) {
  int t = blockIdx.x * 256 + threadIdx.x;
  int total = 9 * 128 * Cin;
  if (t >= total) return;
  int ci = t % Cin;
  int co = (t / Cin) % 128;
  int tap = t / (Cin * 128);
  wp[t] = w[((size_t)co * Cin + ci) * 9 + tap];
}

// ---------------------------------------------------------------------------
// Scalar 3x3 conv (pad=1) for small-channel cases (conv_rec, SpyNet).
// out = addsrc + act(conv + bias)   (addsrc may alias out elementwise)
// ---------------------------------------------------------------------------
__global__ void conv3x3_s(const float* __restrict__ in,
                          const float* __restrict__ Wc,
                          const float* __restrict__ bias,
                          const float* __restrict__ addsrc,
                          float* __restrict__ out, int B, int Cin, int Cout,
                          int H, int Wd, int act) {
  int t = blockIdx.x * 256 + threadIdx.x;
  int total = B * Cout * H * Wd;
  if (t >= total) return;
  int x = t % Wd;
  int y = (t / Wd) % H;
  int co = (t / (Wd * H)) % Cout;
  int b = t / (Wd * H * Cout);
  float acc = bias ? bias[co] : 0.f;
  for (int ci = 0; ci < Cin; ++ci) {
    const float* ip = in + ((size_t)b * Cin + ci) * H * Wd;
    const float* wp = Wc + ((size_t)co * Cin + ci) * 9;
    for (int ky = 0; ky < 3; ++ky) {
      int yy = y + ky - 1;
      if (yy < 0 || yy >= H) continue;
      for (int kx = 0; kx < 3; ++kx) {
        int xx = x + kx - 1;
        if (xx < 0 || xx >= Wd) continue;
        acc += ip[yy * Wd + xx] * wp[ky * 3 + kx];
      }
    }
  }
  acc = actf(acc, act);
  if (addsrc) acc += addsrc[t];
  out[t] = acc;
}

// -------------------------- small utility kernels --------------------------
__global__ void zero_fill(float* p, int n) {
  int t = blockIdx.x * 256 + threadIdx.x;
  if (t < n) p[t] = 0.f;
}

// f1[b=n*6+t] = lq[t][n], f2 uses t+1. lq layout: frame t: (n,3,hw)
__global__ void f12_build(float* __restrict__ F1, float* __restrict__ F2,
                          const float* __restrict__ LQ) {
  int t = blockIdx.x * 256 + threadIdx.x;
  int total = 12 * 3 * kHW;
  if (t >= total) return;
  int p = t % kHW;
  int c = (t / kHW) % 3;
  int bt = t / (kHW * 3);
  int n = bt / 6, tf = bt % 6;
  F1[t] = LQ[((size_t)tf * kN + n) * 3 * kHW + c * kHW + p];
  F2[t] = LQ[((size_t)(tf + 1) * kN + n) * 3 * kHW + c * kHW + p];
}

__global__ void avgpool2(const float* __restrict__ in, float* __restrict__ out,
                         int B, int C, int Ho, int Wo) {
  int t = blockIdx.x * 256 + threadIdx.x;
  int total = B * C * Ho * Wo;
  if (t >= total) return;
  int x = t % Wo, y = (t / Wo) % Ho;
  int c = (t / (Wo * Ho)) % C, b = t / (Wo * Ho * C);
  int Wi = Wo * 2;
  const float* p = in + (((size_t)b * C + c) * (Ho * 2) + 2 * y) * Wi + 2 * x;
  out[t] = 0.25f * (p[0] + p[1] + p[Wi] + p[Wi + 1]);
}

// out(B,2,Ho,Wo) = 2 * bilinear_resize(in(B,2,Ho/2,Wo/2)), half-pixel centers
__global__ void upsample2x(const float* __restrict__ in, float* __restrict__ out,
                           int B, int C, int Ho, int Wo) {
  int t = blockIdx.x * 256 + threadIdx.x;
  int total = B * C * Ho * Wo;
  if (t >= total) return;
  int x = t % Wo, y = (t / Wo) % Ho;
  int c = (t / (Wo * Ho)) % C, b = t / (Wo * Ho * C);
  int Hi = Ho / 2, Wi = Wo / 2;
  float sx = (x + 0.5f) * 0.5f - 0.5f;
  float sy = (y + 0.5f) * 0.5f - 0.5f;
  float fx = floorf(sx), fy = floorf(sy);
  float dx = sx - fx, dy = sy - fy;
  int x0 = clampi((int)fx, 0, Wi - 1), x1 = clampi((int)fx + 1, 0, Wi - 1);
  int y0 = clampi((int)fy, 0, Hi - 1), y1 = clampi((int)fy + 1, 0, Hi - 1);
  const float* p = in + ((size_t)b * C + c) * Hi * Wi;
  float v = (1.f - dx) * (1.f - dy) * p[y0 * Wi + x0] +
            dx * (1.f - dy) * p[y0 * Wi + x1] +
            (1.f - dx) * dy * p[y1 * Wi + x0] + dx * dy * p[y1 * Wi + x1];
  out[t] = 2.f * v;
}

// bilinear_warp of 3-channel img with flow (clip-gather semantics)
__global__ void warp3(const float* __restrict__ img, const float* __restrict__ flow,
                      float* __restrict__ out, int B, int H, int Wd) {
  int t = blockIdx.x * 256 + threadIdx.x;
  int total = B * H * Wd;
  if (t >= total) return;
  int x = t % Wd, y = (t / Wd) % H, b = t / (Wd * H);
  const float* fl = flow + (size_t)b * 2 * H * Wd;
  float sx = x + fl[y * Wd + x];
  float sy = y + fl[H * Wd + y * Wd + x];
  float fx = floorf(sx), fy = floorf(sy);
  float dx = sx - fx, dy = sy - fy;
  int x0 = clampi((int)fx, 0, Wd - 1), x1 = clampi((int)fx + 1, 0, Wd - 1);
  int y0 = clampi((int)fy, 0, H - 1), y1 = clampi((int)fy + 1, 0, H - 1);
  float wa = (1.f - dx) * (1.f - dy), wb = (1.f - dx) * dy;
  float wc = dx * (1.f - dy), wd = dx * dy;
  for (int c = 0; c < 3; ++c) {
    const float* p = img + ((size_t)b * 3 + c) * H * Wd;
    out[((size_t)b * 3 + c) * H * Wd + y * Wd + x] =
        wa * p[y0 * Wd + x0] + wb * p[y1 * Wd + x0] + wc * p[y0 * Wd + x1] +
        wd * p[y1 * Wd + x1];
  }
}

// pack [a(3) | warped(3) | flow(2)] -> 8-channel SpyNet input
__global__ void spy_pack(const float* __restrict__ a, const float* __restrict__ wp,
                         const float* __restrict__ fl, float* __restrict__ out,
                         int B, int H, int Wd) {
  int t = blockIdx.x * 256 + threadIdx.x;
  int total = B * 8 * H * Wd;
  if (t >= total) return;
  int p = t % (H * Wd);
  int ch = (t / (H * Wd)) % 8;
  int b = t / (H * Wd * 8);
  float v;
  if (ch < 3)      v = a[((size_t)b * 3 + ch) * H * Wd + p];
  else if (ch < 6) v = wp[((size_t)b * 3 + (ch - 3)) * H * Wd + p];
  else             v = fl[((size_t)b * 2 + (ch - 6)) * H * Wd + p];
  out[t] = v;
}

// ------------------------------ implicit warp ------------------------------
DEVI float pe_val(int c, float rx, float ry) {
  int g = c >> 5;
  float om = expf(-4.605170186f * (float)(c & 31) * (1.f / 32.f)); // 0.01^(i/32)
  float v = (g < 2 ? rx : ry) * om;
  return (g & 1) ? cosf(v) : sinf(v);
}

__global__ void iw_geom(const float* __restrict__ flows, int idx,
                        float* __restrict__ geom, int* __restrict__ gidx) {
  int t = blockIdx.x * 256 + threadIdx.x;
  if (t >= kN * kHW) return;
  int p = t % kHW, n = t / kHW;
  int y = p >> 6, x = p & 63;
  const float* fl = flows + ((size_t)(n * 6 + idx) * 2) * kHW;
  float sx = x + fl[p];
  float sy = y + fl[kHW + p];
  float fx = floorf(sx), fy = floorf(sy);
  geom[(size_t)t * 2 + 0] = sx - fx;
  geom[(size_t)t * 2 + 1] = sy - fy;
  for (int j = 0; j < 4; ++j) {
    int ox = j & 1, oy = j >> 1;
    int nx = clampi((int)fx + ox, 0, 63);
    int ny = clampi((int)fy + oy, 0, 63);
    gidx[(size_t)t * 4 + j] = ny * 64 + nx;
  }
}

__global__ void iw_build(const float* __restrict__ fprop,
                         const float* __restrict__ fcur,
                         const float* __restrict__ geom,
                         const int* __restrict__ gidx,
                         float* __restrict__ qin, float* __restrict__ kin,
                         float* __restrict__ vin) {
  int t = blockIdx.x * 256 + threadIdx.x;
  if (t >= (int)kNCHW) return;
  int c = t % kC;
  int p = (t / kC) % kHW;
  int n = t / (kC * kHW);
  size_t base = (size_t)n * kHW + p;
  float dx = geom[base * 2 + 0], dy = geom[base * 2 + 1];
  qin[base * kC + c] = fcur[((size_t)n * kC + c) * kHW + p] + pe_val(c, dx, dy);
  for (int j = 0; j < 4; ++j) {
    int ox = j & 1, oy = j >> 1;
    int g = gidx[base * 4 + j];
    float nf = fprop[((size_t)n * kC + c) * kHW + g];
    kin[(base * 4 + j) * kC + c] = nf + pe_val(c, (float)ox - dx, (float)oy - dy);
    vin[(base * 4 + j) * kC + c] = nf;
  }
}

__global__ void iw_attn(const float* __restrict__ Q, const float* __restrict__ K,
                        const float* __restrict__ V, float* __restrict__ O) {
  int t = blockIdx.x * 256 + threadIdx.x;
  if (t >= kN * kHW * kNH) return;
  int h = t % kNH;
  int np = t / kNH;
  const float* q = Q + (size_t)np * kC + h * 16;
  float lg[4];
  float mx = -1e30f;
  for (int j = 0; j < 4; ++j) {
    const float* kk = K + ((size_t)np * 4 + j) * kC + h * 16;
    float s = 0.f;
    for (int d = 0; d < 16; ++d) s += q[d] * kk[d];
    lg[j] = s * 0.25f; // 1/sqrt(16)
    mx = fmaxf(mx, lg[j]);
  }
  float se = 0.f;
  for (int j = 0; j < 4; ++j) { lg[j] = expf(lg[j] - mx); se += lg[j]; }
  float inv = 1.f / se;
  for (int d = 0; d < 16; ++d) {
    float o = 0.f;
    for (int j = 0; j < 4; ++j)
      o += lg[j] * V[((size_t)np * 4 + j) * kC + h * 16 + d];
    O[(size_t)np * kC + h * 16 + d] = o * inv;
  }
}

// xflat[n, tt*HW+p, c] = feats6[n, c, p]   (tt = 0 seed)
__global__ void xflat_nc(float* __restrict__ xflat, const float* __restrict__ f6) {
  int t = blockIdx.x * 256 + threadIdx.x;
  if (t >= (int)kNCHW) return;
  int c = t % kC;
  int p = (t / kC) % kHW;
  int n = t / (kC * kHW);
  xflat[((size_t)n * kL + p) * kC + c] = f6[((size_t)n * kC + c) * kHW + p];
}

// commit attention output to (n,C,hw) prop buffer and xflat slot tt
__global__ void prop_commit(const float* __restrict__ src, float* __restrict__ fprop,
                            float* __restrict__ xflat, int tt) {
  int t = blockIdx.x * 256 + threadIdx.x;
  if (t >= (int)kNCHW) return;
  int c = t % kC;
  int p = (t / kC) % kHW;
  int n = t / (kC * kHW);
  float v = src[((size_t)n * kHW + p) * kC + c];
  fprop[((size_t)n * kC + c) * kHW + p] = v;
  xflat[((size_t)n * kL + tt * kHW + p) * kC + c] = v;
}

// --------------------------------- Mamba -----------------------------------
// depthwise causal width-3 conv over L on xs (= xz[..., :128]) + SiLU
__global__ void mamba_dwconv(const float* __restrict__ xz,
                             const float* __restrict__ cw,
                             const float* __restrict__ cb,
                             float* __restrict__ xs2) {
  int t = blockIdx.x * 256 + threadIdx.x;
  int total = kN * kL * kDI;
  if (t >= total) return;
  int d = t % kDI;
  int l = (t / kDI) % kL;
  int b = t / (kDI * kL);
  const float* xp = xz + (size_t)b * kL * 2 * kDI + d;
  float acc = cb[d];
  for (int k = 0; k < 3; ++k) {
    int lp = l - 2 + k;
    if (lp >= 0) acc += xp[(size_t)lp * 2 * kDI] * cw[d * 3 + k];
  }
  xs2[((size_t)b * kL + l) * kDI + d] = acc / (1.f + expf(-acc)); // silu
}

__global__ void aprep(const float* __restrict__ alog, float* __restrict__ A) {
  int t = blockIdx.x * 256 + threadIdx.x;
  if (t < kDI * kDS) A[t] = -expf(alog[t]);
}

// selective scan: 16 lanes (states) per (b,d) group, shfl reduction for y
__global__ void mamba_scan(const float* __restrict__ xs2,
                           const float* __restrict__ dbc,
                           const float* __restrict__ dtm,
                           const float* __restrict__ A, float* __restrict__ ys) {
  int gid = (blockIdx.x * blockDim.x + threadIdx.x) >> 4;
  int s = threadIdx.x & 15;
  if (gid >= kN * kDI) return;
  int b = gid / kDI, d = gid % kDI;
  float Ads = A[d * kDS + s];
  float h = 0.f;
  for (int l = 0; l < kL; ++l) {
    size_t base = (size_t)b * kL + l;
    float dt = dtm[base * kDI + d];
    float xv = xs2[base * kDI + d];
    float Bv = dbc[base * 40 + 8 + s];
    float Cv = dbc[base * 40 + 24 + s];
    h = expf(dt * Ads) * h + dt * xv * Bv;
    float yv = h * Cv;
    yv += __shfl_xor(yv, 1, 16);
    yv += __shfl_xor(yv, 2, 16);
    yv += __shfl_xor(yv, 4, 16);
    yv += __shfl_xor(yv, 8, 16);
    if (s == 0) ys[base * kDI + d] = yv;
  }
}

// ys = (ys + xs2*D) * silu(z)    (in place on ys)
__global__ void ymix(float* __restrict__ ys, const float* __restrict__ xs2,
                     const float* __restrict__ xz, const float* __restrict__ Dp) {
  int t = blockIdx.x * 256 + threadIdx.x;
  int total = kN * kL * kDI;
  if (t >= total) return;
  int d = t % kDI;
  int l = (t / kDI) % kL;
  int b = t / (kDI * kL);
  size_t base = (size_t)b * kL + l;
  float z = xz[base * 2 * kDI + kDI + d];
  float y = ys[base * kDI + d] + xs2[base * kDI + d] * Dp[d];
  ys[base * kDI + d] = y * (z / (1.f + expf(-z)));
}

// Xp[(b*HW+p), c'] with c' = c128*7 + tt  <-  ym[b, tt*HW+p, c128]
__global__ void xp_pack(const float* __restrict__ ym, float* __restrict__ xp) {
  int t = blockIdx.x * 256 + threadIdx.x;
  int total = kN * kHW * (kC * 7);
  if (t >= total) return;
  int cp = t % (kC * 7);
  int p = (t / (kC * 7)) % kHW;
  int b = t / ((kC * 7) * kHW);
  int c = cp / 7, tt = cp % 7;
  xp[t] = ym[((size_t)b * kL + tt * kHW + p) * kC + c];
}

// (b,p,c) -> (b,c,p)
__global__ void omid_tr(const float* __restrict__ src, float* __restrict__ dst) {
  int t = blockIdx.x * 256 + threadIdx.x;
  if (t >= (int)kNCHW) return;
  int p = t % kHW;
  int c = (t / kHW) % kC;
  int b = t / (kHW * kC);
  dst[((size_t)b * kC + c) * kHW + p] = src[((size_t)b * kHW + p) * kC + c];
}

// ---------------------------------------------------------------------------
static void launch_gemm(const float* X, int ldx, int K, const float* Wt, int ldw,
                        const float* bias, float* Y, int ldy, int R, int Co,
                        int act, hipStream_t s) {
  if (Co <= 48) {
    dim3 g(R / 64, 1);
    gemm_xw<3><<<g, dim3(128), 0, s>>>(X, ldx, K, Wt, ldw, bias, Y, ldy, R, Co,
                                       act);
  } else {
    dim3 g(R / 64, CDIV(Co, 128));
    gemm_xw<8><<<g, dim3(128), 0, s>>>(X, ldx, K, Wt, ldw, bias, Y, ldy, R, Co,
                                       act);
  }
}

extern "C" void kernel_launch(void* const* d_in, const int* in_sizes, int n_in,
                              void* d_out, int out_size, void* d_ws,
                              size_t ws_size, hipStream_t stream) {
  const float* l[7];
  for (int i = 0; i < 7; ++i) l[i] = (const float*)d_in[i];
  const float* conv_rec_w = (const float*)d_in[7];
  const float* conv_rec_b = (const float*)d_in[8];
  const float* conv3_w = (const float*)d_in[9];
  const float* conv3_b = (const float*)d_in[10];
  const float* conv1x1_w = (const float*)d_in[11];
  const float* conv1x1_b = (const float*)d_in[12];
  const float* conv13_w = (const float*)d_in[13];
  const float* conv13_b = (const float*)d_in[14];
  const float* wq_w = (const float*)d_in[15];
  const float* wq_b = (const float*)d_in[16];
  const float* wk_w = (const float*)d_in[17];
  const float* wk_b = (const float*)d_in[18];
  const float* wv_w = (const float*)d_in[19];
  const float* wv_b = (const float*)d_in[20];
  const float* wo_w = (const float*)d_in[21];
  const float* wo_b = (const float*)d_in[22];
  const float* m_in_w = (const float*)d_in[23];
  const float* m_conv_w = (const float*)d_in[24];
  const float* m_conv_b = (const float*)d_in[25];
  const float* m_xproj_w = (const float*)d_in[26];
  const float* m_dt_w = (const float*)d_in[27];
  const float* m_dt_b = (const float*)d_in[28];
  const float* m_Alog = (const float*)d_in[29];
  const float* m_D = (const float*)d_in[30];
  const float* m_out_w = (const float*)d_in[31];
  const float* spy1_w = (const float*)d_in[32];
  const float* spy1_b = (const float*)d_in[33];
  const float* spy2_w = (const float*)d_in[34];
  const float* spy2_b = (const float*)d_in[35];
  float* out = (float*)d_out;
  float* ws = (float*)d_ws;

  // -------------------- workspace layout (float offsets) -------------------
  size_t o = 0;
  auto alloc = [&](size_t n) { size_t r = o; o += n; return r; };
  const size_t FEATS = alloc(7 * kNCHW);
  const size_t LQ    = alloc((size_t)7 * kN * 3 * kHW);
  const size_t F1    = alloc((size_t)12 * 3 * kHW);
  const size_t F2    = alloc((size_t)12 * 3 * kHW);
  const size_t P1A   = alloc((size_t)12 * 3 * 1024);
  const size_t P1B   = alloc((size_t)12 * 3 * 256);
  const size_t P2A   = alloc((size_t)12 * 3 * 1024);
  const size_t P2B   = alloc((size_t)12 * 3 * 256);
  const size_t FLOWA = alloc((size_t)12 * 2 * kHW);
  const size_t FLOWB = alloc((size_t)12 * 2 * kHW);
  const size_t WARPB = alloc((size_t)12 * 3 * kHW);
  const size_t SPYIN = alloc((size_t)12 * 8 * kHW);
  const size_t SPYH  = alloc((size_t)12 * 32 * kHW);
  const size_t GEOM  = alloc((size_t)kN * kHW * 2);
  const size_t GIDXo = alloc((size_t)kN * kHW * 4);
  const size_t QIN   = alloc(kNCHW);
  const size_t QB    = alloc(kNCHW);
  const size_t OBUF  = alloc(kNCHW);
  const size_t PROPN = alloc(kNCHW);
  const size_t FPROP = alloc(kNCHW);
  const size_t VB    = alloc(4 * kNCHW);
  const size_t WP3   = alloc((size_t)9 * 128 * 128);   // packed conv3 weights
  const size_t WP13  = alloc((size_t)9 * 128 * 256);   // packed conv13 weights
  const size_t XFLAT = alloc((size_t)kN * kL * kC);      // later reused as YS
  const size_t XZ    = alloc((size_t)kN * kL * 2 * kDI); // also KIN/VIN/KB; later XP
  const size_t XS2o  = alloc((size_t)kN * kL * kDI);
  const size_t DBCo  = alloc((size_t)kN * kL * 40);
  const size_t DTo   = alloc((size_t)kN * kL * kDI);     // later reused as YM
  const size_t AM    = alloc((size_t)kDI * kDS);
  const size_t OMID  = alloc(kNCHW);
  const size_t OMIDT = alloc(kNCHW);
  (void)ws_size; (void)in_sizes; (void)n_in; (void)out_size;

  float* KIN = ws + XZ;
  float* VIN = ws + XZ + 4 * kNCHW;
  float* KB  = ws + XZ + 8 * kNCHW;
  int* gidx = (int*)(ws + GIDXo);

  // -------------------- weight packing -------------------------------------
  wpack9<<<CDIV(9 * 128 * 128, 256), 256, 0, stream>>>(conv3_w, ws + WP3, 128);
  wpack9<<<CDIV(9 * 128 * 256, 256), 256, 0, stream>>>(conv13_w, ws + WP13, 256);

  // -------------------- Stage A/B: conv_rec & conv3 per frame --------------
  for (int t = 0; t < 7; ++t) {
    conv3x3_s<<<CDIV(kN * 3 * kHW, 256), 256, 0, stream>>>(
        l[t], conv_rec_w, conv_rec_b, nullptr, ws + LQ + (size_t)t * kN * 3 * kHW,
        kN, kC, 3, kH, kW, 0);
    conv3x3_wmma_k<<<dim3(128), dim3(128), 0, stream>>>(
        l[t], kC, l[t], 0, ws + WP3, conv3_b, nullptr,
        ws + FEATS + (size_t)t * kNCHW, 1);
  }

  // -------------------- SpyNet flow ---------------------------------------
  f12_build<<<CDIV(12 * 3 * kHW, 256), 256, 0, stream>>>(ws + F1, ws + F2, ws + LQ);
  avgpool2<<<CDIV(12 * 3 * 1024, 256), 256, 0, stream>>>(ws + F1, ws + P1A, 12, 3, 32, 32);
  avgpool2<<<CDIV(12 * 3 * 256, 256), 256, 0, stream>>>(ws + P1A, ws + P1B, 12, 3, 16, 16);
  avgpool2<<<CDIV(12 * 3 * 1024, 256), 256, 0, stream>>>(ws + F2, ws + P2A, 12, 3, 32, 32);
  avgpool2<<<CDIV(12 * 3 * 256, 256), 256, 0, stream>>>(ws + P2A, ws + P2B, 12, 3, 16, 16);

  const float* As[3] = {ws + P1B, ws + P1A, ws + F1};
  const float* Bs[3] = {ws + P2B, ws + P2A, ws + F2};
  const int hs[3] = {16, 32, 64};
  float* fcur = ws + FLOWA;
  float* falt = ws + FLOWB;
  zero_fill<<<CDIV(12 * 2 * 256, 256), 256, 0, stream>>>(fcur, 12 * 2 * 256);
  for (int li = 0; li < 3; ++li) {
    int h = hs[li], w = h, hw = h * w;
    if (li > 0) {
      upsample2x<<<CDIV(12 * 2 * hw, 256), 256, 0, stream>>>(fcur, falt, 12, 2, h, w);
      float* tmp = fcur; fcur = falt; falt = tmp;
    }
    warp3<<<CDIV(12 * hw, 256), 256, 0, stream>>>(Bs[li], fcur, ws + WARPB, 12, h, w);
    spy_pack<<<CDIV(12 * 8 * hw, 256), 256, 0, stream>>>(As[li], ws + WARPB, fcur,
                                                         ws + SPYIN, 12, h, w);
    conv3x3_s<<<CDIV(12 * 32 * hw, 256), 256, 0, stream>>>(
        ws + SPYIN, spy1_w, spy1_b, nullptr, ws + SPYH, 12, 8, 32, h, w, 1);
    conv3x3_s<<<CDIV(12 * 2 * hw, 256), 256, 0, stream>>>(
        ws + SPYH, spy2_w, spy2_b, fcur, fcur, 12, 32, 2, h, w, 0);
  }
  const float* flows = fcur; // (n*6+t, 2, 64, 64)

  // -------------------- implicit-warp recurrence ---------------------------
  xflat_nc<<<CDIV((int)kNCHW, 256), 256, 0, stream>>>(ws + XFLAT,
                                                      ws + FEATS + 6 * kNCHW);
  const float* fprop = ws + FEATS + 6 * kNCHW;
  for (int i = 1; i <= 6; ++i) {
    int idx = 6 - i;
    iw_geom<<<CDIV(kN * kHW, 256), 256, 0, stream>>>(flows, idx, ws + GEOM, gidx);
    iw_build<<<CDIV((int)kNCHW, 256), 256, 0, stream>>>(
        fprop, ws + FEATS + (size_t)idx * kNCHW, ws + GEOM, gidx, ws + QIN, KIN, VIN);
    launch_gemm(ws + QIN, kC, kC, wq_w, kC, wq_b, ws + QB, kC, kN * kHW, kC, 0, stream);
    launch_gemm(KIN, kC, kC, wk_w, kC, wk_b, KB, kC, kN * kHW * 4, kC, 0, stream);
    launch_gemm(VIN, kC, kC, wv_w, kC, wv_b, ws + VB, kC, kN * kHW * 4, kC, 0, stream);
    iw_attn<<<CDIV(kN * kHW * kNH, 256), 256, 0, stream>>>(ws + QB, KB, ws + VB,
                                                           ws + OBUF);
    launch_gemm(ws + OBUF, kC, kC, wo_w, kC, wo_b, ws + PROPN, kC, kN * kHW, kC, 0,
                stream);
    prop_commit<<<CDIV((int)kNCHW, 256), 256, 0, stream>>>(ws + PROPN, ws + FPROP,
                                                           ws + XFLAT, i);
    fprop = ws + FPROP;
  }

  // -------------------- Mamba ---------------------------------------------
  launch_gemm(ws + XFLAT, kC, kC, m_in_w, kC, nullptr, ws + XZ, 2 * kDI, kN * kL,
              2 * kDI, 0, stream);
  mamba_dwconv<<<CDIV(kN * kL * kDI, 256), 256, 0, stream>>>(ws + XZ, m_conv_w,
                                                             m_conv_b, ws + XS2o);
  launch_gemm(ws + XS2o, kDI, kDI, m_xproj_w, kDI, nullptr, ws + DBCo, 40, kN * kL,
              40, 0, stream);
  launch_gemm(ws + DBCo, 40, 8, m_dt_w, 8, m_dt_b, ws + DTo, kDI, kN * kL, kDI, 2,
              stream);
  aprep<<<CDIV(kDI * kDS, 256), 256, 0, stream>>>(m_Alog, ws + AM);
  mamba_scan<<<16, 256, 0, stream>>>(ws + XS2o, ws + DBCo, ws + DTo, ws + AM,
                                     ws + XFLAT /* ys */);
  ymix<<<CDIV(kN * kL * kDI, 256), 256, 0, stream>>>(ws + XFLAT, ws + XS2o, ws + XZ,
                                                     m_D);
  launch_gemm(ws + XFLAT, kDI, kDI, m_out_w, kDI, nullptr, ws + DTo /* ym */, kC,
              kN * kL, kC, 0, stream);

  // -------------------- 1x1 fusion conv ------------------------------------
  xp_pack<<<CDIV(kN * kHW * kC * 7, 256), 256, 0, stream>>>(ws + DTo,
                                                            ws + XZ /* Xp */);
  launch_gemm(ws + XZ, kC * 7, kC * 7, conv1x1_w, kC * 7, conv1x1_b, ws + OMID, kC,
              kN * kHW, kC, 1, stream);
  omid_tr<<<CDIV((int)kNCHW, 256), 256, 0, stream>>>(ws + OMID, ws + OMIDT);

  // -------------------- conv13 + residual per frame ------------------------
  for (int t = 0; t < 7; ++t) {
    conv3x3_wmma_k<<<dim3(128), dim3(128), 0, stream>>>(
        ws + FEATS + (size_t)t * kNCHW, kC, ws + OMIDT, kC, ws + WP13, conv13_b,
        l[t], out + (size_t)t * kNCHW, 1);
  }
}